// CopyDecoder_69853348102591
// MI455X (gfx1250) — compile-verified
//
#include <hip/hip_runtime.h>
#include <hip/hip_bf16.h>
#include <math.h>

// Sizes from the reference
#define Vv 50257
#define Ee 1024
#define Hh 1024
#define Ss 2048
#define NPROB (Vv + Ss)          // 52305
#define OUT_H1 NPROB             // h1 at out[52305..53328]
#define OUT_C1 (NPROB + Hh)      // c1 at out[53329..54352]

typedef __attribute__((ext_vector_type(2))) float v2f;
typedef __attribute__((ext_vector_type(4))) float v4f;
typedef __attribute__((ext_vector_type(8))) float v8f;

// ---------------------------------------------------------------------------
// ws layout (floats)
//   0     xh[2048]            (concat(x, h0) for attention GEMV)
//   2048  rho[2048]
//   4096  ctx[2048]
//   6144  selctx[2048]
//   8192  att[2048]           (logits -> softmax in place)
//   10240 lstm_in[3072]       ([x | selective | attentive])
//   13312 gates[4096]
//   17408 sc_part[4*2048]     (score_c N-split partials)
// ---------------------------------------------------------------------------
#define WS_XH 0
#define WS_RHO 2048
#define WS_CTX 4096
#define WS_SEL 6144
#define WS_ATT 8192
#define WS_LIN 10240
#define WS_GATES 13312
#define WS_SCPART 17408

// --------------------------- pack x|h0 and lstm_in[0:1024] ------------------
__global__ void prep_kernel(const float* __restrict__ x,
                            const float* __restrict__ h0,
                            float* __restrict__ xh,
                            float* __restrict__ lin) {
  int i = blockIdx.x * blockDim.x + threadIdx.x;
  if (i < 1024) { xh[i] = x[i]; lin[i] = x[i]; }
  else if (i < 2048) { xh[i] = h0[i - 1024]; }
}

// --------------------------- generic row GEMV: y = W x + b ------------------
// One wave per row (wave32), 16B loads, shuffle reduction. Bandwidth-bound.
// NT=true streams W with non-temporal hints (for the 205 MB Wo projection,
// which would otherwise evict enc/Wc from the 192 MB L2).
template <bool NT>
__global__ void gemv_rows(const float* __restrict__ W, const float* __restrict__ x,
                          const float* __restrict__ b, float* __restrict__ y,
                          int R, int C) {
  extern __shared__ float s_x[];
  for (int c = threadIdx.x; c < C; c += blockDim.x) s_x[c] = x[c];
  __syncthreads();
  const int wave = threadIdx.x >> 5, lane = threadIdx.x & 31;
  const int wpb = blockDim.x >> 5;
  for (int row = blockIdx.x * wpb + wave; row < R; row += gridDim.x * wpb) {
    const float* wr = W + (size_t)row * C;
    float sum = 0.f;
    for (int c = lane * 4; c < C; c += 128) {
      v4f wv;
      if (NT) wv = __builtin_nontemporal_load((const v4f*)(wr + c));
      else    wv = *(const v4f*)(wr + c);
      v4f xv = *(const v4f*)(s_x + c);
      sum += wv.x * xv.x + wv.y * xv.y + wv.z * xv.z + wv.w * xv.w;
    }
#pragma unroll
    for (int o = 16; o > 0; o >>= 1) sum += __shfl_down(sum, o, 32);
    if (lane == 0) y[row] = sum + b[row];
  }
}

// --------------------------- rho (copy-attention weights) -------------------
__global__ void rho_kernel(const int* __restrict__ sentence,
                           const float* __restrict__ prev_probs,
                           const int* __restrict__ prev_word,
                           float* __restrict__ rho) {
  __shared__ float red[8];
  const int pw = prev_word[0];
  const int lane = threadIdx.x & 31, wv = threadIdx.x >> 5;
  float s = 0.f;
  for (int i = threadIdx.x; i < Ss; i += blockDim.x) {
    float v = (sentence[i] == pw) ? prev_probs[Vv + i] : 0.f;
    rho[i] = v;
    s += v;
  }
#pragma unroll
  for (int o = 16; o > 0; o >>= 1) s += __shfl_xor(s, o, 32);
  if (lane == 0) red[wv] = s;
  __syncthreads();
  if (threadIdx.x == 0) {
    float t = 0.f;
    for (int w = 0; w < 8; ++w) t += red[w];
    red[0] = 1.0f / (t + 1e-9f);
  }
  __syncthreads();
  const float inv = red[0];
  for (int i = threadIdx.x; i < Ss; i += blockDim.x) rho[i] *= inv;
}

// --------------------------- single-block softmax (in place) ----------------
__global__ void softmax1b(float* __restrict__ p, int n) {
  __shared__ float red[32];
  __shared__ float bcast;
  const int tid = threadIdx.x, nt = blockDim.x;
  const int lane = tid & 31, wv = tid >> 5, nw = nt >> 5;
  float m = -3.402823466e38f;
  for (int i = tid; i < n; i += nt) m = fmaxf(m, p[i]);
#pragma unroll
  for (int o = 16; o > 0; o >>= 1) m = fmaxf(m, __shfl_xor(m, o, 32));
  if (lane == 0) red[wv] = m;
  __syncthreads();
  if (tid == 0) {
    float t = red[0];
    for (int w = 1; w < nw; ++w) t = fmaxf(t, red[w]);
    bcast = t;
  }
  __syncthreads();
  m = bcast;
  float s = 0.f;
  for (int i = tid; i < n; i += nt) {
    float e = __expf(p[i] - m);
    p[i] = e;
    s += e;
  }
#pragma unroll
  for (int o = 16; o > 0; o >>= 1) s += __shfl_xor(s, o, 32);
  __syncthreads();
  if (lane == 0) red[wv] = s;
  __syncthreads();
  if (tid == 0) {
    float t = 0.f;
    for (int w = 0; w < nw; ++w) t += red[w];
    bcast = 1.0f / t;
  }
  __syncthreads();
  const float inv = bcast;
  for (int i = tid; i < n; i += nt) p[i] *= inv;
}

// ------------- fused column GEMVs: ctx = w1@enc, selctx = w2@enc ------------
// Reads the 16 MB encoder matrix exactly once for both reductions.
__global__ void dual_colgemv(const float* __restrict__ enc,
                             const float* __restrict__ w1,
                             const float* __restrict__ w2,
                             float* __restrict__ o1, float* __restrict__ o2) {
  __shared__ float s1[256], s2[256];
  const int j = blockIdx.x * 256 + threadIdx.x;
  float a1 = 0.f, a2 = 0.f;
  for (int i0 = 0; i0 < Ss; i0 += 256) {
    __syncthreads();
    s1[threadIdx.x] = w1[i0 + threadIdx.x];
    s2[threadIdx.x] = w2[i0 + threadIdx.x];
    __syncthreads();
#pragma unroll 8
    for (int ii = 0; ii < 256; ++ii) {
      float e = enc[(size_t)(i0 + ii) * 2048 + j];
      a1 += s1[ii] * e;
      a2 += s2[ii] * e;
    }
  }
  o1[j] = a1;
  o2[j] = a2;
}

// --------------------------- LSTM gate GEMV (two mats fused) ----------------
__global__ void lstm_gates(const float* __restrict__ Wih, const float* __restrict__ Whh,
                           const float* __restrict__ bih, const float* __restrict__ bhh,
                           const float* __restrict__ lin, const float* __restrict__ h0,
                           float* __restrict__ gates) {
  __shared__ float s[4096];
  for (int c = threadIdx.x; c < 3072; c += blockDim.x) s[c] = lin[c];
  for (int c = threadIdx.x; c < 1024; c += blockDim.x) s[3072 + c] = h0[c];
  __syncthreads();
  const int wave = threadIdx.x >> 5, lane = threadIdx.x & 31;
  const int wpb = blockDim.x >> 5;
  for (int row = blockIdx.x * wpb + wave; row < 4096; row += gridDim.x * wpb) {
    const float* w1 = Wih + (size_t)row * 3072;
    const float* w2 = Whh + (size_t)row * 1024;
    float sum = 0.f;
    for (int c = lane * 4; c < 3072; c += 128) {
      v4f wv = *(const v4f*)(w1 + c);
      v4f xv = *(const v4f*)(s + c);
      sum += wv.x * xv.x + wv.y * xv.y + wv.z * xv.z + wv.w * xv.w;
    }
    for (int c = lane * 4; c < 1024; c += 128) {
      v4f wv = *(const v4f*)(w2 + c);
      v4f xv = *(const v4f*)(s + 3072 + c);
      sum += wv.x * xv.x + wv.y * xv.y + wv.z * xv.z + wv.w * xv.w;
    }
#pragma unroll
    for (int o = 16; o > 0; o >>= 1) sum += __shfl_down(sum, o, 32);
    if (lane == 0) gates[row] = sum + bih[row] + bhh[row];
  }
}

// --------------------------- LSTM pointwise update --------------------------
__global__ void lstm_update(const float* __restrict__ gates,
                            const float* __restrict__ c0,
                            float* __restrict__ out) {
  const int r = threadIdx.x; // 1024 threads
  const float ig = gates[r], fg = gates[1024 + r], gg = gates[2048 + r], og = gates[3072 + r];
  const float si = 1.f / (1.f + __expf(-ig));
  const float sf = 1.f / (1.f + __expf(-fg));
  const float so = 1.f / (1.f + __expf(-og));
  const float c1 = sf * c0[r] + si * tanhf(gg);
  const float h1 = so * tanhf(c1);
  out[OUT_H1 + r] = h1;
  out[OUT_C1 + r] = c1;
}

// ------- score_c: fused  sum_j tanh(enc @ Wc.T + b)[i,j] * h1[j]  -----------
// The only compute-dense op (8.6 GFLOP fp32): V_WMMA_F32_16X16X4_F32.
// Block = 16 enc rows (M). grid.y splits the 64 N-tiles into 4 groups; each
// of the 8 waves owns TWO 16-wide N-tiles with persistent accumulators.
// The shared A tile (16 rows x 512-float K-chunk) is staged into LDS with
// CDNA5 async global->LDS B128 loads (ASYNCcnt), then每 wave reads A once via
// ds_load and feeds two WMMAs -> halves A-side vector-memory traffic.
// fp32 A/B lane layout (ISA 7.12.2): lanes 0-15 hold K={0,1}, 16-31 K={2,3}.
#define KC 512
#define ASTR (KC + 2)  // +2 floats per row: break 512 % 64-bank aliasing

__global__ void scorec_wmma(const float* __restrict__ enc,
                            const float* __restrict__ Wc,
                            const float* __restrict__ Wcb,
                            const float* __restrict__ h1,
                            float* __restrict__ sc_part) {
  __shared__ float s_a[16 * ASTR];
  __shared__ float s_part[8 * 16];

  const int i0 = blockIdx.x * 16;
  const int wave = threadIdx.x >> 5;
  const int lane = threadIdx.x & 31;
  const int half = lane >> 4;   // selects K pair {0,1} vs {2,3}
  const int l16 = lane & 15;    // M (for A) / N (for B) within tile

  // Two N-tiles per wave within this block's group of 16 tiles.
  const int j0a = (blockIdx.y * 16 + wave) << 4;
  const int j0b = j0a + 128;
  const float* browa = Wc + (size_t)(j0a + l16) * 2048 + (half << 1);
  const float* browb = Wc + (size_t)(j0b + l16) * 2048 + (half << 1);

  v8f acca = {}, accb = {};

  for (int kc = 0; kc < 2048; kc += KC) {
    __syncthreads();  // previous chunk fully consumed
    // ---- async-stage A chunk: 16 rows x KC floats, 16B per lane-op --------
#pragma unroll
    for (int it = 0; it < 8; ++it) {
      const int idx = threadIdx.x + it * 256;  // float4 index in chunk
      const int row = idx >> 7;                // / (KC/4)
      const int c4 = idx & 127;
      unsigned laddr = (unsigned)(size_t)(&s_a[row * ASTR + c4 * 4]);
      const float* g = enc + (size_t)(i0 + row) * 2048 + kc + c4 * 4;
      asm volatile("global_load_async_to_lds_b128 %0, %1, off"
                   :: "v"(laddr), "v"(g) : "memory");
    }
    asm volatile("s_wait_asynccnt 0x0" ::: "memory");
    __syncthreads();

    const float* al = &s_a[l16 * ASTR + (half << 1)];
#pragma unroll 4
    for (int k = 0; k < KC; k += 4) {
      v2f a = *(const v2f*)(al + k);
      v2f b0 = *(const v2f*)(browa + kc + k);
      acca = __builtin_amdgcn_wmma_f32_16x16x4_f32(false, a, false, b0,
                                                   (short)0, acca, false, false);
      v2f b1 = *(const v2f*)(browb + kc + k);
      accb = __builtin_amdgcn_wmma_f32_16x16x4_f32(false, a, false, b1,
                                                   (short)0, accb, false, false);
    }
  }

  // D layout: element v of this lane is (M = half*8 + v, N = j0 + l16).
  const int ja = j0a + l16, jb = j0b + l16;
  const float biasa = Wcb[ja], ha = h1[ja];
  const float biasb = Wcb[jb], hb = h1[jb];
  float partial[8];
#pragma unroll
  for (int v = 0; v < 8; ++v)
    partial[v] = tanhf(acca[v] + biasa) * ha + tanhf(accb[v] + biasb) * hb;

  // Reduce over the 16 lanes of each half (sums over N), deterministic.
#pragma unroll
  for (int v = 0; v < 8; ++v) {
    float t = partial[v];
    t += __shfl_xor(t, 1, 32);
    t += __shfl_xor(t, 2, 32);
    t += __shfl_xor(t, 4, 32);
    t += __shfl_xor(t, 8, 32);
    partial[v] = t;
  }
  if (l16 == 0) {
#pragma unroll
    for (int v = 0; v < 8; ++v) s_part[wave * 16 + (half << 3) + v] = partial[v];
  }
  __syncthreads();
  if (threadIdx.x < 16) {
    float s = 0.f;
    for (int w = 0; w < 8; ++w) s += s_part[w * 16 + threadIdx.x];
    sc_part[blockIdx.y * 2048 + i0 + threadIdx.x] = s;
  }
}

__global__ void scorec_reduce(const float* __restrict__ part, float* __restrict__ out) {
  const int i = blockIdx.x * blockDim.x + threadIdx.x;
  if (i < Ss) out[i] = part[i] + part[2048 + i] + part[4096 + i] + part[6144 + i];
}

// ---------------------------------------------------------------------------
extern "C" void kernel_launch(void* const* d_in, const int* in_sizes, int n_in,
                              void* d_out, int out_size, void* d_ws, size_t ws_size,
                              hipStream_t stream) {
  const float* x      = (const float*)d_in[0];
  const float* enc    = (const float*)d_in[1];
  const int*   sent   = (const int*)d_in[2];
  const float* pprobs = (const float*)d_in[3];
  const float* h0     = (const float*)d_in[4];
  const float* c0     = (const float*)d_in[5];
  const float* attn_W = (const float*)d_in[6];
  const float* attn_b = (const float*)d_in[7];
  const float* comb_W = (const float*)d_in[8];
  const float* comb_b = (const float*)d_in[9];
  const float* Ws_W   = (const float*)d_in[10];
  const float* Ws_b   = (const float*)d_in[11];
  const float* Wo_W   = (const float*)d_in[12];
  const float* Wo_b   = (const float*)d_in[13];
  const float* Wc_W   = (const float*)d_in[14];
  const float* Wc_b   = (const float*)d_in[15];
  const float* W_ih   = (const float*)d_in[16];
  const float* W_hh   = (const float*)d_in[17];
  const float* b_ih   = (const float*)d_in[18];
  const float* b_hh   = (const float*)d_in[19];
  const int*   pword  = (const int*)d_in[20];

  float* out = (float*)d_out;
  float* ws  = (float*)d_ws;
  float* xh     = ws + WS_XH;
  float* rho    = ws + WS_RHO;
  float* ctx    = ws + WS_CTX;
  float* selctx = ws + WS_SEL;
  float* att    = ws + WS_ATT;
  float* lin    = ws + WS_LIN;
  float* gates  = ws + WS_GATES;
  float* scpart = ws + WS_SCPART;

  // 1. pack [x | h0] and lstm_in[0:1024] = x
  prep_kernel<<<8, 256, 0, stream>>>(x, h0, xh, lin);
  // 2. attention logits: att = attn_W @ [x|h] + b   (16 MB, bandwidth)
  gemv_rows<false><<<256, 256, 2048 * sizeof(float), stream>>>(attn_W, xh, attn_b, att, Ss, 2048);
  // 3. rho (copy mechanism weights), normalized
  rho_kernel<<<1, 256, 0, stream>>>(sent, pprobs, pword, rho);
  // 4. softmax over attention logits (in place)
  softmax1b<<<1, 256, 0, stream>>>(att, Ss);
  // 5. one pass over enc for BOTH ctx = attn@enc and selctx = rho@enc
  dual_colgemv<<<8, 256, 0, stream>>>(enc, att, rho, ctx, selctx);
  // 6./7. attentive -> lstm_in[2048:], selective -> lstm_in[1024:]
  gemv_rows<false><<<128, 256, 2048 * sizeof(float), stream>>>(comb_W, ctx, comb_b, lin + 2048, Hh, 2048);
  gemv_rows<false><<<128, 256, 2048 * sizeof(float), stream>>>(Ws_W, selctx, Ws_b, lin + 1024, Hh, 2048);
  // 8. gate preactivations (fused W_ih + W_hh GEMVs, 64 MB)
  lstm_gates<<<512, 256, 0, stream>>>(W_ih, W_hh, b_ih, b_hh, lin, h0, gates);
  // 9. LSTM pointwise -> h1, c1 straight into d_out
  lstm_update<<<1, 1024, 0, stream>>>(gates, c0, out);
  // 10. score_g = Wo @ h1 + b  (205 MB streamed non-temporal: the bandwidth
  //     floor of this step; NT keeps enc/Wc resident in the 192 MB L2)
  gemv_rows<true><<<1024, 256, 1024 * sizeof(float), stream>>>(Wo_W, out + OUT_H1, Wo_b, out, Vv, 1024);
  // 11. score_c via fp32 WMMA, async-LDS-staged A tile, fused tanh/h1 epilogue
  scorec_wmma<<<dim3(128, 4), 256, 0, stream>>>(enc, Wc_W, Wc_b, out + OUT_H1, scpart);
  scorec_reduce<<<8, 256, 0, stream>>>(scpart, out + Vv);
  // 12. softmax over concat(score_g, score_c)
  softmax1b<<<1, 1024, 0, stream>>>(out, NPROB);
}